// SR_Loss_29643864277362
// MI455X (gfx1250) — compile-verified
//
#include <hip/hip_runtime.h>
#include <hip/hip_bf16.h>
#include <math.h>

typedef __attribute__((ext_vector_type(2))) float v2f;
typedef __attribute__((ext_vector_type(8))) float v8f;

#define N_OBJ 8192
#define N_SR  4096
#define N_V   4100
#define N_F   4096
#define EPSF  1e-8f
#define FACE_SLICES 8

// ---------------------------------------------------------------------------
// Prep: pad coords to float4 (zero-padded K slot feeds the 16x16x4 WMMA
// directly), precompute |sr|^2 per point (keeps the NN inner loop free of
// LDS shuffles), build per-face edge-cross vectors + scalar constants.
// ---------------------------------------------------------------------------
__global__ void prep_kernel(const float* __restrict__ obj,
                            const float* __restrict__ sr,
                            const float* __restrict__ hv,
                            const float* __restrict__ fn,
                            const int*   __restrict__ hf,
                            float*  __restrict__ obj4,
                            float*  __restrict__ sr4,
                            float*  __restrict__ srNorm,    // |sr|^2
                            float*  __restrict__ faceVec,   // [N_F][16]: n,0,c0,0,c1,0,c2,0
                            float4* __restrict__ faceConst, // {n.v0, c0.v0, c1.v1, c2.v2}
                            int*    __restrict__ hitCount,
                            float*  __restrict__ accum)
{
    const int t = blockIdx.x * blockDim.x + threadIdx.x;
    if (t < N_OBJ) {
        obj4[t*4+0] = obj[t*3+0];
        obj4[t*4+1] = obj[t*3+1];
        obj4[t*4+2] = obj[t*3+2];
        obj4[t*4+3] = 0.0f;
        hitCount[t] = 0;
    }
    if (t < N_SR) {
        const float x = sr[t*3+0], y = sr[t*3+1], z = sr[t*3+2];
        sr4[t*4+0] = x;
        sr4[t*4+1] = y;
        sr4[t*4+2] = z;
        sr4[t*4+3] = 0.0f;            // pad MUST stay 0 (WMMA K=4 slot)
        srNorm[t]  = x*x + y*y + z*z;
    }
    if (t < N_F) {
        const int i0 = hf[t*3+0], i1 = hf[t*3+1], i2 = hf[t*3+2];
        const float v0x = hv[i0*3+0], v0y = hv[i0*3+1], v0z = hv[i0*3+2];
        const float v1x = hv[i1*3+0], v1y = hv[i1*3+1], v1z = hv[i1*3+2];
        const float v2x = hv[i2*3+0], v2y = hv[i2*3+1], v2z = hv[i2*3+2];
        const float nx  = fn[t*3+0],  ny  = fn[t*3+1],  nz  = fn[t*3+2];

        // c = cross(n, vb - va) for edges (v0,v1), (v1,v2), (v2,v0)
        float e0x = v1x-v0x, e0y = v1y-v0y, e0z = v1z-v0z;
        float e1x = v2x-v1x, e1y = v2y-v1y, e1z = v2z-v1z;
        float e2x = v0x-v2x, e2y = v0y-v2y, e2z = v0z-v2z;
        float c0x = ny*e0z - nz*e0y, c0y = nz*e0x - nx*e0z, c0z = nx*e0y - ny*e0x;
        float c1x = ny*e1z - nz*e1y, c1y = nz*e1x - nx*e1z, c1z = nx*e1y - ny*e1x;
        float c2x = ny*e2z - nz*e2y, c2y = nz*e2x - nx*e2z, c2z = nx*e2y - ny*e2x;

        float* fv = faceVec + t*16;
        fv[0]=nx;   fv[1]=ny;   fv[2]=nz;   fv[3]=0.0f;
        fv[4]=c0x;  fv[5]=c0y;  fv[6]=c0z;  fv[7]=0.0f;
        fv[8]=c1x;  fv[9]=c1y;  fv[10]=c1z; fv[11]=0.0f;
        fv[12]=c2x; fv[13]=c2y; fv[14]=c2z; fv[15]=0.0f;

        faceConst[t] = make_float4(nx*v0x + ny*v0y + nz*v0z,
                                   c0x*v0x + c0y*v0y + c0z*v0z,
                                   c1x*v1x + c1y*v1y + c1z*v1z,
                                   c2x*v2x + c2y*v2y + c2z*v2z);
    }
    if (t == 0) *accum = 0.0f;
}

// ---------------------------------------------------------------------------
// Stage 1: fused NN argmin via WMMA f32 16x16x4 (K=3 padded to 4).
// One wave = one 16-obj tile; sweeps all 256 sr tiles.
// d2 = |obj|^2 + |sr|^2 - 2 obj.sr ; |obj|^2 is constant per row -> add at end.
// Inner loop: 1 b64 load + 1 b32 load + 1 WMMA + 8 fma + 8 cmp/select pairs,
// no LDS traffic (norms precomputed).
// ---------------------------------------------------------------------------
__global__ __launch_bounds__(32)
void nn_kernel(const float* __restrict__ obj4,
               const float* __restrict__ sr4,
               const float* __restrict__ srNorm,
               float* __restrict__ D4,       // padded ray vectors trg - obj
               float* __restrict__ cmap_out) // d_out + 1
{
    const int lane = threadIdx.x & 31;
    const int half = lane >> 4;
    const int sub  = lane & 15;
    const int kb   = half * 2;          // K base: 0 or 2
    const int objTile = blockIdx.x;

    // A fragment: 16x4 f32, lane row = sub, K = kb..kb+1 (pad slot is 0)
    v2f afrag;
    {
        const int o = objTile*16 + sub;
        afrag.x = obj4[o*4 + kb];
        afrag.y = obj4[o*4 + kb + 1];
    }

    float best[8];
    int   bidx[8];
#pragma unroll
    for (int i = 0; i < 8; ++i) { best[i] = 3.4e38f; bidx[i] = 0; }

    for (int st = 0; st < N_SR/16; ++st) {
        const int s = st*16 + sub;
        v2f b;
        b.x = sr4[s*4 + kb];
        b.y = sr4[s*4 + kb + 1];
        const float sq = srNorm[s];          // |sr_col|^2, same value in both halves

        v8f acc = {};
        acc = __builtin_amdgcn_wmma_f32_16x16x4_f32(false, afrag, false, b,
                                                    (short)0, acc, false, false);
#pragma unroll
        for (int i = 0; i < 8; ++i) {
            float v = fmaf(-2.0f, acc[i], sq);   // |sr|^2 - 2 obj.sr
            if (v < best[i]) { best[i] = v; bidx[i] = s; }
        }
    }

    // Reduce 16 column-slots per row through LDS.
    __shared__ float sbv[16][16];
    __shared__ int   sbi[16][16];
#pragma unroll
    for (int i = 0; i < 8; ++i) {
        const int m = i + 8*half;
        sbv[m][sub] = best[i];
        sbi[m][sub] = bidx[i];
    }
    __syncthreads();

    if (lane < 16) {
        const int m = lane;
        float bv = sbv[m][0];
        int   bi = sbi[m][0];
#pragma unroll
        for (int j = 1; j < 16; ++j) {
            if (sbv[m][j] < bv) { bv = sbv[m][j]; bi = sbi[m][j]; }
        }
        const int o = objTile*16 + m;
        const float ox = obj4[o*4+0], oy = obj4[o*4+1], oz = obj4[o*4+2];
        float d2 = bv + ox*ox + oy*oy + oz*oz;
        d2 = fmaxf(d2, 0.0f);
        const float nn = sqrtf(d2);
        // cmap = 1 - 2*(sigmoid(100 d) - 0.5) = 2 / (1 + exp(100 d))
        cmap_out[o] = 2.0f / (1.0f + __expf(100.0f * nn));

        const float sx = sr4[bi*4+0], sy = sr4[bi*4+1], sz = sr4[bi*4+2];
        D4[o*4+0] = sx - ox;
        D4[o*4+1] = sy - oy;
        D4[o*4+2] = sz - oz;
        D4[o*4+3] = 0.0f;
    }
}

// ---------------------------------------------------------------------------
// Stage 2: ray-triangle parity. Per (16-face x 16-obj) tile issue 8 WMMAs
// (A in {n,c0,c1,c2} x B in {obj,D}); all 8 result tiles share the same
// (vgpr,lane)->(face,point) mapping so the hit predicate is evaluated
// entirely in registers on the VALU, co-executing with the XDL matrix pipe.
// ---------------------------------------------------------------------------
__global__ __launch_bounds__(32)
void hit_kernel(const float*  __restrict__ obj4,
                const float*  __restrict__ D4,
                const float*  __restrict__ faceVec,
                const float4* __restrict__ faceConst,
                int* __restrict__ hitCount)
{
    const int lane = threadIdx.x & 31;
    const int half = lane >> 4;
    const int sub  = lane & 15;
    const int kb   = half * 2;
    const int objTile = blockIdx.x;
    const int slice   = blockIdx.y;
    const int tilesPerSlice = (N_F/16) / FACE_SLICES;

    // B fragments for this obj tile (fixed across the face sweep)
    const int oc = objTile*16 + sub;
    v2f bo, bd;
    bo.x = obj4[oc*4 + kb]; bo.y = obj4[oc*4 + kb + 1];
    bd.x = D4[oc*4 + kb];   bd.y = D4[oc*4 + kb + 1];

    int cnt = 0;
    for (int ft = slice*tilesPerSlice; ft < (slice+1)*tilesPerSlice; ++ft) {
        const float* fv = faceVec + (size_t)(ft*16 + sub) * 16;
        v2f an, ac0, ac1, ac2;
        an.x  = fv[kb];      an.y  = fv[kb+1];
        ac0.x = fv[4+kb];    ac0.y = fv[4+kb+1];
        ac1.x = fv[8+kb];    ac1.y = fv[8+kb+1];
        ac2.x = fv[12+kb];   ac2.y = fv[12+kb+1];

        const v8f z = {};
        v8f nD  = __builtin_amdgcn_wmma_f32_16x16x4_f32(false, an,  false, bd, (short)0, z, false, false);
        v8f nO  = __builtin_amdgcn_wmma_f32_16x16x4_f32(false, an,  false, bo, (short)0, z, false, false);
        v8f c0O = __builtin_amdgcn_wmma_f32_16x16x4_f32(false, ac0, false, bo, (short)0, z, false, false);
        v8f c0D = __builtin_amdgcn_wmma_f32_16x16x4_f32(false, ac0, false, bd, (short)0, z, false, false);
        v8f c1O = __builtin_amdgcn_wmma_f32_16x16x4_f32(false, ac1, false, bo, (short)0, z, false, false);
        v8f c1D = __builtin_amdgcn_wmma_f32_16x16x4_f32(false, ac1, false, bd, (short)0, z, false, false);
        v8f c2O = __builtin_amdgcn_wmma_f32_16x16x4_f32(false, ac2, false, bo, (short)0, z, false, false);
        v8f c2D = __builtin_amdgcn_wmma_f32_16x16x4_f32(false, ac2, false, bd, (short)0, z, false, false);

#pragma unroll
        for (int i = 0; i < 8; ++i) {
            const int m = ft*16 + i + 8*half;          // face index of this C row
            const float4 fc = faceConst[m];
            const float denom = nD[i];
            const bool  dm = fabsf(denom) > EPSF;
            const float t  = (fc.x - nO[i]) / (dm ? denom : 1.0f);
            const bool e0 = fmaf(t, c0D[i], c0O[i] - fc.y) >= 0.0f;
            const bool e1 = fmaf(t, c1D[i], c1O[i] - fc.z) >= 0.0f;
            const bool e2 = fmaf(t, c2D[i], c2O[i] - fc.w) >= 0.0f;
            cnt += (e0 & e1 & e2 & dm & (t > EPSF)) ? 1 : 0;
        }
    }
    // lane L and L+16 cover the same obj column (different face rows)
    cnt += __shfl_xor(cnt, 16, 32);
    if (lane < 16) atomicAdd(&hitCount[objTile*16 + lane], cnt);
}

// ---------------------------------------------------------------------------
// pen_dist = sqrt( sum_{interior} |D|^2 + 1e-12 )
// ---------------------------------------------------------------------------
__global__ void pen_accum_kernel(const int* __restrict__ hitCount,
                                 const float* __restrict__ D4,
                                 float* __restrict__ accum)
{
    const int t = blockIdx.x * blockDim.x + threadIdx.x;
    if (t < N_OBJ && (hitCount[t] & 1)) {
        const float dx = D4[t*4+0], dy = D4[t*4+1], dz = D4[t*4+2];
        atomicAdd(accum, dx*dx + dy*dy + dz*dz);
    }
}

__global__ void pen_write_kernel(const float* __restrict__ accum,
                                 float* __restrict__ out0)
{
    if (threadIdx.x == 0 && blockIdx.x == 0)
        out0[0] = sqrtf(*accum + 1e-12f);
}

// ---------------------------------------------------------------------------
extern "C" void kernel_launch(void* const* d_in, const int* in_sizes, int n_in,
                              void* d_out, int out_size, void* d_ws, size_t ws_size,
                              hipStream_t stream)
{
    const float* obj = (const float*)d_in[0];  // 8192 x 3
    const float* sr  = (const float*)d_in[1];  // 4096 x 3
    const float* hv  = (const float*)d_in[2];  // 4100 x 3
    const float* fn  = (const float*)d_in[3];  // 4096 x 3
    const int*   hf  = (const int*)  d_in[4];  // 4096 x 3

    char* ws = (char*)d_ws;
    size_t off = 0;
    float*  obj4      = (float*)(ws + off); off += (size_t)N_OBJ * 4 * sizeof(float);
    float*  sr4       = (float*)(ws + off); off += (size_t)N_SR  * 4 * sizeof(float);
    float*  srNorm    = (float*)(ws + off); off += (size_t)N_SR  * sizeof(float);
    float*  D4        = (float*)(ws + off); off += (size_t)N_OBJ * 4 * sizeof(float);
    float*  faceVec   = (float*)(ws + off); off += (size_t)N_F  * 16 * sizeof(float);
    float4* faceConst = (float4*)(ws + off); off += (size_t)N_F * sizeof(float4);
    int*    hitCount  = (int*)  (ws + off); off += (size_t)N_OBJ * sizeof(int);
    float*  accum     = (float*)(ws + off); off += sizeof(float);

    float* out = (float*)d_out;          // [0] = pen_dist, [1..8192] = cmap

    prep_kernel<<<(N_OBJ + 255)/256, 256, 0, stream>>>(
        obj, sr, hv, fn, hf, obj4, sr4, srNorm, faceVec, faceConst, hitCount, accum);

    nn_kernel<<<N_OBJ/16, 32, 0, stream>>>(obj4, sr4, srNorm, D4, out + 1);

    hit_kernel<<<dim3(N_OBJ/16, FACE_SLICES), 32, 0, stream>>>(
        obj4, D4, faceVec, faceConst, hitCount);

    pen_accum_kernel<<<(N_OBJ + 255)/256, 256, 0, stream>>>(hitCount, D4, accum);
    pen_write_kernel<<<1, 32, 0, stream>>>(accum, out);
}